// RelAttention_68831145886324
// MI455X (gfx1250) — compile-verified
//
#include <hip/hip_runtime.h>
#include <hip/hip_bf16.h>
#include <math.h>

// ---------------------------------------------------------------------------
// Problem constants (from reference): B=32, F_IN=8192, F_OUT=16384, HEADS=512
// ---------------------------------------------------------------------------
#define BATCH   32
#define F_IN    8192
#define F_OUT   16384
#define HEADS   512

typedef __attribute__((ext_vector_type(2))) float v2f;
typedef __attribute__((ext_vector_type(4))) float v4f;
typedef __attribute__((ext_vector_type(8))) float v8f;
typedef __attribute__((ext_vector_type(4))) int   v4i;

// K-chunk staged through LDS per iteration
#define KC      128
// padded LDS row stride (floats): +4 -> 4-bank rotation per row, conflict-free
// b64 A-fragment reads; 132 % 4 == 0 keeps v4f/b128 destinations 16B-aligned
#define XS      (KC + 4)

// ---- CDNA5 async global->LDS path (ASYNCcnt), with portable fallback ------
#if __has_builtin(__builtin_amdgcn_global_load_async_to_lds_b128)
#define HAS_ASYNC_LDS 1
typedef __attribute__((address_space(1))) v4i gv4i;   // global int4
typedef __attribute__((address_space(3))) v4i lv4i;   // LDS int4
#else
#define HAS_ASYNC_LDS 0
#endif

#if __has_builtin(__builtin_amdgcn_s_wait_asynccnt)
#define WAIT_ASYNC(n) __builtin_amdgcn_s_wait_asynccnt(n)
#elif HAS_ASYNC_LDS
#define WAIT_ASYNC(n) asm volatile("s_wait_asynccnt %0" ::"i"(n) : "memory")
#else
#define WAIT_ASYNC(n) ((void)0)
#endif

// exact-erf GELU (torch nn.GELU default)
__device__ __forceinline__ float gelu_erf(float z) {
    return 0.5f * z * (1.0f + erff(z * 0.70710678118654752f));
}

// ---------------------------------------------------------------------------
// Kernel 1: fused QKV projection + bias + GELU, via V_WMMA_F32_16X16X4_F32.
//   out[m, n] = gelu( sum_k x[m,k] * W[n,k] + b[n] )
// One wave computes a 32x32 output tile (four 16x16 WMMA accumulators);
// the two A fragments (x rows 0..15 / 16..31) are shared by both N-tiles.
// x chunks are double-buffered in LDS and filled with async global->LDS
// (GLOBAL_LOAD_ASYNC_TO_LDS_B128, ASYNCcnt) overlapping the WMMA stream.
// blockDim = 128 (4 waves) -> 128 output columns per block.
// blockIdx.y selects which of {Wq,Wk,Wv} / {bq,bk,bv} / {q,k,v}.
// W traffic is stream-once => non-temporal loads keep x/q/k/v cache-resident.
// ---------------------------------------------------------------------------
__global__ void __launch_bounds__(128)
qkv_gemm_gelu_kernel(const float* __restrict__ x,
                     const float* __restrict__ Wq, const float* __restrict__ Wk,
                     const float* __restrict__ Wv,
                     const float* __restrict__ bq, const float* __restrict__ bk,
                     const float* __restrict__ bv,
                     float* __restrict__ qo, float* __restrict__ ko,
                     float* __restrict__ vo) {
    const int sel = blockIdx.y;
    const float* __restrict__ W = (sel == 0) ? Wq : (sel == 1) ? Wk : Wv;
    const float* __restrict__ bias = (sel == 0) ? bq : (sel == 1) ? bk : bv;
    float* __restrict__ out = (sel == 0) ? qo : (sel == 1) ? ko : vo;

    __shared__ float ldsx[2 * 32 * XS];   // ~33.8 KB: double-buffered x chunks

    const int tid  = threadIdx.x;
    const int wave = tid >> 5;        // 0..3
    const int lane = tid & 31;
    const int lo   = lane & 15;       // M row / N col within tile
    const int hi   = lane >> 4;       // K sub-pair selector (A/B layout)

    const int nbase = (blockIdx.x * 4 + wave) * 32;       // wave's 32 columns
    const int ncolA = nbase + lo;
    const int ncolB = nbase + 16 + lo;
    // B fragment source: contiguous f32 pair W[ncol][k + 2*hi .. +1]
    const float* __restrict__ wrowA = W + (size_t)ncolA * F_IN + 2 * hi;
    const float* __restrict__ wrowB = W + (size_t)ncolB * F_IN + 2 * hi;

    // stage x[0:32][kc:kc+KC] into LDS buffer `buf` (8 x b128 per thread)
    auto issue_chunk = [&](int kc, int buf) {
        float* dstbase = ldsx + buf * (32 * XS);
#pragma unroll
        for (int i = 0; i < 8; ++i) {
            const int flat = i * 128 + tid;       // 0..1023 v4f chunks
            const int row  = flat >> 5;           // 0..31
            const int c4   = (flat & 31) << 2;    // 0,4,...,124
            const float* src = x + (size_t)row * F_IN + kc + c4;
            float* dst = dstbase + row * XS + c4;
#if HAS_ASYNC_LDS
            __builtin_amdgcn_global_load_async_to_lds_b128(
                (gv4i*)src, (lv4i*)dst, /*offset=*/0, /*cpol=*/0);
#else
            *(v4f*)dst = *(const v4f*)src;
#endif
        }
    };

    v8f c00 = {};  // rows  0..15, cols nbase+ 0..15
    v8f c01 = {};  // rows 16..31, cols nbase+ 0..15
    v8f c10 = {};  // rows  0..15, cols nbase+16..31
    v8f c11 = {};  // rows 16..31, cols nbase+16..31

    // prologue: start filling buffer 0 with chunk 0
    issue_chunk(0, 0);

    for (int kc = 0; kc < F_IN; kc += KC) {
        const int buf = (kc / KC) & 1;
        const float* cur = ldsx + buf * (32 * XS);

        if (kc + KC < F_IN) {
            // overlap: fill the other buffer while we compute on this one
            issue_chunk(kc + KC, buf ^ 1);
            // pull next chunk's weight lines too (global_prefetch)
            __builtin_prefetch(wrowA + kc + KC, 0, 0);
            __builtin_prefetch(wrowB + kc + KC, 0, 0);
            WAIT_ASYNC(8);   // wait only for the older 8 in-flight fills
        } else {
            WAIT_ASYNC(0);
        }
        __syncthreads();     // chunk `buf` visible to all waves

#pragma unroll 8
        for (int kk = 0; kk < KC; kk += 4) {
            // A fragments (16x4 f32): lane lo = row, f32 pair at k = kk+2*hi
            v2f a0 = *(const v2f*)&cur[lo * XS + kk + 2 * hi];
            v2f a1 = *(const v2f*)&cur[(16 + lo) * XS + kk + 2 * hi];
            // B fragments (4x16 f32): stream-once -> non-temporal
            v2f bA = __builtin_nontemporal_load(
                (const v2f*)(wrowA + kc + kk));
            v2f bB = __builtin_nontemporal_load(
                (const v2f*)(wrowB + kc + kk));

            c00 = __builtin_amdgcn_wmma_f32_16x16x4_f32(
                false, a0, false, bA, (short)0, c00, false, false);
            c01 = __builtin_amdgcn_wmma_f32_16x16x4_f32(
                false, a1, false, bA, (short)0, c01, false, false);
            c10 = __builtin_amdgcn_wmma_f32_16x16x4_f32(
                false, a0, false, bB, (short)0, c10, false, false);
            c11 = __builtin_amdgcn_wmma_f32_16x16x4_f32(
                false, a1, false, bB, (short)0, c11, false, false);
        }
        __syncthreads();     // all waves done reading `buf`; safe to refill
    }

    // epilogue: bias + erf-GELU + store.
    // C/D layout: lane lo = N col; VGPR r holds M = r + 8*hi (+16 for *1 tiles)
    const float bnA = bias[ncolA];
    const float bnB = bias[ncolB];
#pragma unroll
    for (int r = 0; r < 8; ++r) {
        const int m = r + 8 * hi;
        out[(size_t)m * F_OUT + ncolA]        = gelu_erf(c00[r] + bnA);
        out[(size_t)(m + 16) * F_OUT + ncolA] = gelu_erf(c01[r] + bnA);
        out[(size_t)m * F_OUT + ncolB]        = gelu_erf(c10[r] + bnB);
        out[(size_t)(m + 16) * F_OUT + ncolB] = gelu_erf(c11[r] + bnB);
    }
}

// ---------------------------------------------------------------------------
// Kernel 2: heads-axis softmax attention.
// One block per (b, i); 512 threads, thread = head h.
//   s[h,j] = q[b,h,i]*k[b,h,j]/sqrt(32) + tw[rel(i,j)]
//   a      = softmax over h (block-wide reduction, wave32 shuffles + LDS tree)
//   out[b,h,i] = sum_j a[h,j] * v[b,h,j]
// ---------------------------------------------------------------------------
__device__ __forceinline__ float wave_max32(float v) {
#pragma unroll
    for (int off = 16; off > 0; off >>= 1)
        v = fmaxf(v, __shfl_xor(v, off, 32));
    return v;
}
__device__ __forceinline__ float wave_sum32(float v) {
#pragma unroll
    for (int off = 16; off > 0; off >>= 1)
        v += __shfl_xor(v, off, 32);
    return v;
}

__device__ __forceinline__ float block_max512(float v, float* red, int lane, int wid) {
    v = wave_max32(v);
    if (lane == 0) red[wid] = v;
    __syncthreads();
    if (wid == 0) {
        float t = (lane < 16) ? red[lane] : -3.4e38f;
        t = wave_max32(t);
        if (lane == 0) red[0] = t;
    }
    __syncthreads();
    float r = red[0];
    __syncthreads();
    return r;
}
__device__ __forceinline__ float block_sum512(float v, float* red, int lane, int wid) {
    v = wave_sum32(v);
    if (lane == 0) red[wid] = v;
    __syncthreads();
    if (wid == 0) {
        float t = (lane < 16) ? red[lane] : 0.0f;
        t = wave_sum32(t);
        if (lane == 0) red[0] = t;
    }
    __syncthreads();
    float r = red[0];
    __syncthreads();
    return r;
}

__global__ void __launch_bounds__(512)
heads_softmax_attn_kernel(const float* __restrict__ q,
                          const float* __restrict__ k,
                          const float* __restrict__ v,
                          const float* __restrict__ tw,
                          float* __restrict__ out) {
    __shared__ float red[16];

    const int b = blockIdx.x;       // 0..31
    const int i = blockIdx.y;       // 0..31
    const int h = threadIdx.x;      // 0..511
    const int lane = h & 31;
    const int wid  = h >> 5;

    const size_t base = (size_t)b * F_OUT + (size_t)h * 32;
    const float qhi = q[base + i];
    const float inv_sqrt32 = 0.17677669529663689f;

    float acc = 0.0f;
    for (int j = 0; j < 32; ++j) {
        const int d = j - i;
        const float bj = tw[(d >= 0) ? (31 - d) : (30 - d)];
        const float s = qhi * k[base + j] * inv_sqrt32 + bj;

        const float m  = block_max512(s, red, lane, wid);
        const float e  = expf(s - m);
        const float sm = block_sum512(e, red, lane, wid);
        acc += (e / sm) * v[base + j];
    }
    out[base + i] = acc;   // (B, H, 32, 1) flat == b*H*32 + h*32 + i
}

// ---------------------------------------------------------------------------
// Launcher
// ---------------------------------------------------------------------------
extern "C" void kernel_launch(void* const* d_in, const int* in_sizes, int n_in,
                              void* d_out, int out_size, void* d_ws, size_t ws_size,
                              hipStream_t stream) {
    (void)in_sizes; (void)n_in; (void)out_size; (void)ws_size;

    const float* x  = (const float*)d_in[0];
    const float* Wq = (const float*)d_in[1];
    const float* bq = (const float*)d_in[2];
    const float* Wk = (const float*)d_in[3];
    const float* bk = (const float*)d_in[4];
    const float* Wv = (const float*)d_in[5];
    const float* bv = (const float*)d_in[6];
    const float* tw = (const float*)d_in[7];

    float* q = (float*)d_ws;                       // 32*16384 f32 = 2 MB
    float* k = q + (size_t)BATCH * F_OUT;
    float* v = k + (size_t)BATCH * F_OUT;          // total 6 MB scratch

    // 512 tiles of 32 cols / 4 waves per block = 128 blocks; y selects Q/K/V
    dim3 g1(F_OUT / (32 * 4), 3);
    qkv_gemm_gelu_kernel<<<g1, 128, 0, stream>>>(x, Wq, Wk, Wv, bq, bk, bv,
                                                 q, k, v);

    dim3 g2(BATCH, 32);
    heads_softmax_attn_kernel<<<g2, 512, 0, stream>>>(q, k, v, tw,
                                                      (float*)d_out);
}